// DirectBiVoxGO_5669356832913
// MI455X (gfx1250) — compile-verified
//
#include <hip/hip_runtime.h>

typedef float v2f __attribute__((ext_vector_type(2)));
typedef float v8f __attribute__((ext_vector_type(8)));

#define GRID_N   256
#define NSTEPS   256
#define G3       (GRID_N * GRID_N * GRID_N)

// z-adjacent corner pair as one 64-bit load (4-byte alignment is legal for
// non-formatted b64 in DWORD alignment mode), then lerp along z.
__device__ __forceinline__ float zpair_lerp(const float* __restrict__ g,
                                            int idx, float fz) {
    const float2 v = *(const float2*)(g + idx);
    return fmaf(fz, v.y - v.x, v.x);
}

// One block per ray, one thread per step.
// Cumprod along the ray is done in log space; the 256-long cumsum is computed
// as a 16x16 matmul against an upper-triangular ones matrix using
// V_WMMA_F32_16X16X4_F32 (4 chained k-slices) on wave 0.
__global__ __launch_bounds__(256) void render_ray_kernel(
    const float* __restrict__ density,    // [G^3]
    const float* __restrict__ k0,         // [3 * G^3]
    const float* __restrict__ ray_pts,    // [n_rays * 256 * 3]
    const float* __restrict__ act_shift_p,
    const float* __restrict__ interval_p,
    float* __restrict__ out_rgb,          // [n_rays * 3]
    float* __restrict__ out_ainv,         // [n_rays]
    float* __restrict__ out_w)            // [n_rays * 256]
{
    __shared__ float s_la[NSTEPS];     // log(1-alpha) per step (WMMA A input)
    __shared__ float s_scan[NSTEPS];   // within-row inclusive prefix (WMMA D)
    __shared__ float s_rowoff[16];     // exclusive row offsets
    __shared__ float s_part[8][3];     // per-wave color partials
    __shared__ float s_ainv;

    const int ray  = blockIdx.x;
    const int t    = threadIdx.x;
    const int lane = t & 31;
    const int wv   = t >> 5;

    const float act_shift = act_shift_p[0];
    const float interval  = interval_p[0];

    // ---- streamed (read-once) point load: non-temporal so grids stay in L2 ----
    const float* p = ray_pts + ((size_t)ray * NSTEPS + t) * 3;
    const float px = __builtin_nontemporal_load(p + 0);
    const float py = __builtin_nontemporal_load(p + 1);
    const float pz = __builtin_nontemporal_load(p + 2);

    // ---- trilinear setup (align_corners=True), shared by density + k0 ----
    const float scale = 0.5f * (float)(GRID_N - 1);
    const float cx = (px + 1.0f) * scale;
    const float cy = (py + 1.0f) * scale;
    const float cz = (pz + 1.0f) * scale;
    float f0 = fminf(fmaxf(floorf(cx), 0.0f), (float)(GRID_N - 2));
    float f1 = fminf(fmaxf(floorf(cy), 0.0f), (float)(GRID_N - 2));
    float f2 = fminf(fmaxf(floorf(cz), 0.0f), (float)(GRID_N - 2));
    const int bx = (int)f0, by = (int)f1, bz = (int)f2;
    const float fx = cx - f0, fy = cy - f1, fz = cz - f2;

    // 4 (x,y) corner-pair base indices; z handled by the b64 pair lerp
    const int idx00 = (bx * GRID_N + by) * GRID_N + bz;
    const int idx01 = idx00 + GRID_N;             // (bx,   by+1)
    const int idx10 = idx00 + GRID_N * GRID_N;    // (bx+1, by  )
    const int idx11 = idx10 + GRID_N;             // (bx+1, by+1)
    const float w00 = (1.0f - fx) * (1.0f - fy);
    const float w01 = (1.0f - fx) * fy;
    const float w10 = fx * (1.0f - fy);
    const float w11 = fx * fy;

    // ---- 16 x b64 gathers feed all 4 channels (index math shared) ----
    const float* k1 = k0 + G3;
    const float* k2 = k0 + 2 * G3;
    float d  = w00 * zpair_lerp(density, idx00, fz);
    float r0 = w00 * zpair_lerp(k0,      idx00, fz);
    float r1 = w00 * zpair_lerp(k1,      idx00, fz);
    float r2 = w00 * zpair_lerp(k2,      idx00, fz);
    d  = fmaf(w01, zpair_lerp(density, idx01, fz), d);
    r0 = fmaf(w01, zpair_lerp(k0,      idx01, fz), r0);
    r1 = fmaf(w01, zpair_lerp(k1,      idx01, fz), r1);
    r2 = fmaf(w01, zpair_lerp(k2,      idx01, fz), r2);
    d  = fmaf(w10, zpair_lerp(density, idx10, fz), d);
    r0 = fmaf(w10, zpair_lerp(k0,      idx10, fz), r0);
    r1 = fmaf(w10, zpair_lerp(k1,      idx10, fz), r1);
    r2 = fmaf(w10, zpair_lerp(k2,      idx10, fz), r2);
    d  = fmaf(w11, zpair_lerp(density, idx11, fz), d);
    r0 = fmaf(w11, zpair_lerp(k0,      idx11, fz), r0);
    r1 = fmaf(w11, zpair_lerp(k1,      idx11, fz), r1);
    r2 = fmaf(w11, zpair_lerp(k2,      idx11, fz), r2);

    // ---- log-space alpha: log(1-alpha) = -interval * softplus(d + shift) ----
    const float x  = d + act_shift;
    const float sp = fmaxf(x, 0.0f) + __logf(1.0f + __expf(-fabsf(x)));
    const float la = -interval * sp;          // log(1 - alpha), exact, <= 0
    const float alpha = -expm1f(la);          // 1 - exp(la), in [0,1)

    // ---- 256-long cumsum of la via WMMA matmul-scan --------------------------
    s_la[t] = la;
    __syncthreads();

    if (t < 32) {                             // wave 0, EXEC all ones
        const int m  = lane & 15;             // A-matrix row M
        const int kk = (lane >> 4) << 1;      // K pair base: 0 (lanes 0-15) / 2 (16-31)
        const int n  = lane & 15;             // B/D column N
        v8f acc = {0.f, 0.f, 0.f, 0.f, 0.f, 0.f, 0.f, 0.f};
#pragma unroll
        for (int kb = 0; kb < 4; ++kb) {      // K = 16 split into 4 WMMA k-slices
            v2f a, b;
            a.x = s_la[16 * m + 4 * kb + kk];
            a.y = s_la[16 * m + 4 * kb + kk + 1];
            // upper-triangular ones: U[k][n] = (k <= n)
            b.x = ((4 * kb + kk)     <= n) ? 1.0f : 0.0f;
            b.y = ((4 * kb + kk + 1) <= n) ? 1.0f : 0.0f;
            acc = __builtin_amdgcn_wmma_f32_16x16x4_f32(
                false, a, false, b, (short)0, acc, false, false);
        }
        // D layout: VGPR v -> M = v (lanes 0-15) / v+8 (lanes 16-31), N = lane&15
        const int mbase = (lane < 16) ? 0 : 8;
        const int col   = lane & 15;
#pragma unroll
        for (int v = 0; v < 8; ++v)
            s_scan[(mbase + v) * 16 + col] = acc[v];

        // Row totals already live in registers: lane 15 holds rows 0-7
        // (acc[v] = S[v][15]), lane 31 holds rows 8-15. Serial 8-reg scan +
        // one shuffle gives the 16 exclusive row offsets without LDS loops.
        float rt[8];
#pragma unroll
        for (int v = 0; v < 8; ++v) rt[v] = acc[v];
#pragma unroll
        for (int v = 1; v < 8; ++v) rt[v] += rt[v - 1];
        const float tot15 = __shfl(rt[7], 15, 32);   // sum of rows 0-7
        if (lane == 31) {
#pragma unroll
            for (int v = 0; v < 8; ++v) rt[v] += tot15;
        }
        if (lane == 15) {
            s_rowoff[0] = 0.0f;
#pragma unroll
            for (int v = 0; v < 7; ++v) s_rowoff[v + 1] = rt[v];
            s_rowoff[8] = rt[7];
        }
        if (lane == 31) {
#pragma unroll
            for (int v = 0; v < 7; ++v) s_rowoff[v + 9] = rt[v];
        }
    }
    __syncthreads();

    const float cum  = s_scan[t] + s_rowoff[t >> 4]; // inclusive cumsum of la
    const float Texc = __expf(cum - la);      // exclusive cumprod of (1-alpha)
    const float wt   = alpha * Texc;          // weights[ray][t]

    __builtin_nontemporal_store(wt, out_w + (size_t)ray * NSTEPS + t);

    if (t == NSTEPS - 1) {
        const float ainv = __expf(cum);       // Tcum[:, -1]
        s_ainv = ainv;
        out_ainv[ray] = ainv;
    }

    // ---- rgb = sigmoid(k0 interp); reduce weights*rgb over steps -------------
    float a0 = wt * (1.0f / (1.0f + __expf(-r0)));
    float a1 = wt * (1.0f / (1.0f + __expf(-r1)));
    float a2 = wt * (1.0f / (1.0f + __expf(-r2)));
#pragma unroll
    for (int o = 16; o > 0; o >>= 1) {        // wave32 reduction
        a0 += __shfl_down(a0, o, 32);
        a1 += __shfl_down(a1, o, 32);
        a2 += __shfl_down(a2, o, 32);
    }
    if (lane == 0) {
        s_part[wv][0] = a0; s_part[wv][1] = a1; s_part[wv][2] = a2;
    }
    __syncthreads();

    if (t == 0) {
        float o0 = 0.f, o1 = 0.f, o2 = 0.f;
#pragma unroll
        for (int w = 0; w < 8; ++w) {
            o0 += s_part[w][0]; o1 += s_part[w][1]; o2 += s_part[w][2];
        }
        const float ai = s_ainv;
        out_rgb[ray * 3 + 0] = o0 + ai;
        out_rgb[ray * 3 + 1] = o1 + ai;
        out_rgb[ray * 3 + 2] = o2 + ai;
    }
}

extern "C" void kernel_launch(void* const* d_in, const int* in_sizes, int n_in,
                              void* d_out, int out_size, void* d_ws, size_t ws_size,
                              hipStream_t stream) {
    const float* density   = (const float*)d_in[0];
    const float* k0        = (const float*)d_in[1];
    const float* ray_pts   = (const float*)d_in[2];
    const float* act_shift = (const float*)d_in[3];
    const float* interval  = (const float*)d_in[4];

    const int n_rays = in_sizes[2] / (NSTEPS * 3);

    float* out      = (float*)d_out;
    float* out_rgb  = out;                          // [n_rays * 3]
    float* out_ainv = out + (size_t)n_rays * 3;     // [n_rays]
    float* out_w    = out + (size_t)n_rays * 4;     // [n_rays * 256]

    render_ray_kernel<<<n_rays, NSTEPS, 0, stream>>>(
        density, k0, ray_pts, act_shift, interval, out_rgb, out_ainv, out_w);
}